// GraphConvolution_8117488189773
// MI455X (gfx1250) — compile-verified
//
#include <hip/hip_runtime.h>

typedef __attribute__((ext_vector_type(2))) float v2f;
typedef __attribute__((ext_vector_type(8))) float v8f;

#define NNODES 100000
#define NEDGES 3200000
#define DIM    256
#define XW_BYTES (100000L * 256L * 4L)   // 102,400,000 (256-byte aligned)

// ---------------------------------------------------------------------------
// Zero the accumulator/output buffer (float4 stores).
// ---------------------------------------------------------------------------
__global__ __launch_bounds__(256) void zero_kernel(float* __restrict__ out, int n4) {
  int i = blockIdx.x * blockDim.x + threadIdx.x;
  if (i < n4) {
    ((float4*)out)[i] = make_float4(0.f, 0.f, 0.f, 0.f);
  }
}

// ---------------------------------------------------------------------------
// Pack W [256x256] into WMMA B-fragment order so the GEMM inner loop can use
// contiguous b64 loads. Fragment f = ks*16 + nt (ks = K/4 step, nt = 16-col
// tile). Lane l of the fragment holds {W[kA][n], W[kA+1][n]} with
// kA = 4*ks + 2*(l>>4), n = nt*16 + (l&15)  -> stored at Wpk[f*64 + 2*l].
// Total: 64 ksteps * 16 tiles * 64 floats = 65536 floats = 256 KB.
// ---------------------------------------------------------------------------
__global__ __launch_bounds__(256) void pack_w_kernel(const float* __restrict__ W,
                                                     float* __restrict__ Wpk) {
  const int gid  = blockIdx.x * 256 + threadIdx.x;   // 32768 threads total
  const int lane = gid & 31;
  const int nt   = (gid >> 5) & 15;
  const int ks   = gid >> 9;                         // 0..63
  const int l16  = lane & 15;
  const int hi   = lane >> 4;
  const int kA   = ks * 4 + 2 * hi;
  const int n    = nt * 16 + l16;

  v2f b;
  b.x = W[kA * DIM + n];
  b.y = W[(kA + 1) * DIM + n];
  ((v2f*)Wpk)[(ks * 16 + nt) * 32 + lane] = b;
}

// ---------------------------------------------------------------------------
// XW = X @ W using native-precision V_WMMA_F32_16X16X4_F32.
// Block = 128 threads = 4 waves; block computes a 16-row x 256-col strip.
// Wave w owns columns [w*64, w*64+64) as four 16x16 f32 accumulators.
// A fragment: contiguous float2 of this lane's X row (b64 load).
// B fragments: contiguous float2 from the packed Wpk buffer (b64 loads).
// ---------------------------------------------------------------------------
__global__ __launch_bounds__(128) void gemm_xw_kernel(const float* __restrict__ X,
                                                      const float* __restrict__ Wpk,
                                                      float* __restrict__ XW) {
  const int tid  = threadIdx.x;
  const int wave = tid >> 5;
  const int lane = tid & 31;
  const int l16  = lane & 15;
  const int hi   = lane >> 4;              // 0 for lanes 0-15, 1 for 16-31
  const long row0 = (long)blockIdx.x * 16; // 16-row strip
  const int  n0   = wave * 64;             // 64-column strip per wave

  v8f acc0 = {}; v8f acc1 = {}; v8f acc2 = {}; v8f acc3 = {};

  // A: this lane's X row viewed as float2; kstep ks needs v2f index 2*ks + hi
  const v2f* __restrict__ xrow2 = (const v2f*)(X + (row0 + l16) * DIM);
  // B: packed fragments; wave's first tile is nt = wave*4; each fragment = 32 v2f
  const v2f* __restrict__ wbase = (const v2f*)Wpk + (wave * 4) * 32 + lane;

  #pragma unroll 4
  for (int ks = 0; ks < DIM / 4; ++ks) {
    v2f a = xrow2[2 * ks + hi];
    const v2f* wf = wbase + ks * (16 * 32);
    v2f b0 = wf[0 * 32];
    v2f b1 = wf[1 * 32];
    v2f b2 = wf[2 * 32];
    v2f b3 = wf[3 * 32];
    acc0 = __builtin_amdgcn_wmma_f32_16x16x4_f32(false, a, false, b0, (short)0, acc0, false, false);
    acc1 = __builtin_amdgcn_wmma_f32_16x16x4_f32(false, a, false, b1, (short)0, acc1, false, false);
    acc2 = __builtin_amdgcn_wmma_f32_16x16x4_f32(false, a, false, b2, (short)0, acc2, false, false);
    acc3 = __builtin_amdgcn_wmma_f32_16x16x4_f32(false, a, false, b3, (short)0, acc3, false, false);
  }

  // Store D tiles: VGPR v holds row (v + 8*hi), column l16 of each subtile.
  float* __restrict__ obase = XW + row0 * DIM + n0 + l16;
  #pragma unroll
  for (int v = 0; v < 8; ++v) {
    const long m = v + 8 * hi;
    obase[m * DIM +  0] = acc0[v];
    obase[m * DIM + 16] = acc1[v];
    obase[m * DIM + 32] = acc2[v];
    obase[m * DIM + 48] = acc3[v];
  }
}

// ---------------------------------------------------------------------------
// Edge scatter: out[row] += val * XW[col]   (per-column hardware f32 atomics).
// 64 threads per edge, 4 consecutive columns per thread (float4 gather).
// XW (102 MB) and out (102 MB) both fit in the 192 MB L2, so this phase runs
// at L2 bandwidth, not HBM.
// ---------------------------------------------------------------------------
__global__ __launch_bounds__(256) void scatter_kernel(const float* __restrict__ XW,
                                                      const int*   __restrict__ erow,
                                                      const int*   __restrict__ ecol,
                                                      const float* __restrict__ eval,
                                                      float*       __restrict__ out) {
  const long t = (long)blockIdx.x * 256 + threadIdx.x;
  const long e = t >> 6;                 // 64 lanes per edge
  const int  c = (int)(t & 63) << 2;     // 4 columns per lane
  if (e >= NEDGES) return;

  const int   r = erow[e];
  const int   s = ecol[e];
  const float v = eval[e];

  const float4 x4 = *(const float4*)(XW + (long)s * DIM + c);
  float* o = out + (long)r * DIM + c;
  unsafeAtomicAdd(o + 0, v * x4.x);
  unsafeAtomicAdd(o + 1, v * x4.y);
  unsafeAtomicAdd(o + 2, v * x4.z);
  unsafeAtomicAdd(o + 3, v * x4.w);
}

// ---------------------------------------------------------------------------
// In-place ReLU.
// ---------------------------------------------------------------------------
__global__ __launch_bounds__(256) void relu_kernel(float* __restrict__ out, int n4) {
  int i = blockIdx.x * blockDim.x + threadIdx.x;
  if (i < n4) {
    float4 v = ((float4*)out)[i];
    v.x = fmaxf(v.x, 0.f);
    v.y = fmaxf(v.y, 0.f);
    v.z = fmaxf(v.z, 0.f);
    v.w = fmaxf(v.w, 0.f);
    ((float4*)out)[i] = v;
  }
}

extern "C" void kernel_launch(void* const* d_in, const int* in_sizes, int n_in,
                              void* d_out, int out_size, void* d_ws, size_t ws_size,
                              hipStream_t stream) {
  const float* x    = (const float*)d_in[0];   // [NNODES, DIM]
  const float* w    = (const float*)d_in[1];   // [DIM, DIM]
  const int*   erow = (const int*)  d_in[2];   // [NEDGES]
  const int*   ecol = (const int*)  d_in[3];   // [NEDGES]
  const float* ev   = (const float*)d_in[4];   // [NEDGES]
  float*       out  = (float*)d_out;           // [NNODES, DIM]

  float* xw  = (float*)d_ws;                           // 102.4 MB
  float* wpk = (float*)((char*)d_ws + XW_BYTES);       // +256 KB packed W

  const int n4 = NNODES * DIM / 4;             // 6,400,000 float4s

  // 1) zero accumulator (also clears harness poison)
  zero_kernel<<<(n4 + 255) / 256, 256, 0, stream>>>(out, n4);

  // 2) pack W into WMMA B-fragment layout (one-shot, 256 KB)
  pack_w_kernel<<<128, 256, 0, stream>>>(w, wpk);

  // 3) XW = X @ W  (fp32 WMMA), one block per 16-row strip
  gemm_xw_kernel<<<NNODES / 16, 128, 0, stream>>>(x, wpk, xw);

  // 4) COO scatter with hardware f32 atomics: 64 lanes/edge
  const long sthreads = (long)NEDGES * 64;     // 204.8M threads
  scatter_kernel<<<(int)(sthreads / 256), 256, 0, stream>>>(xw, erow, ecol, ev, out);

  // 5) ReLU in place
  relu_kernel<<<(n4 + 255) / 256, 256, 0, stream>>>(out, n4);
}